// GCNLayer_4303557230928
// MI455X (gfx1250) — compile-verified
//
#include <hip/hip_runtime.h>

#define D 64  // D_IN == D_OUT == 64

typedef float v2f __attribute__((ext_vector_type(2)));
typedef float v8f __attribute__((ext_vector_type(8)));

// ---------------------------------------------------------------------------
// Kernel 1: zero the aggregation buffer (d_ws is poisoned, not pre-zeroed)
// ---------------------------------------------------------------------------
__global__ void gcn_zero_kernel(float4* __restrict__ p, int n4) {
    int i = blockIdx.x * blockDim.x + threadIdx.x;
    if (i < n4) p[i] = make_float4(0.f, 0.f, 0.f, 0.f);
}

// ---------------------------------------------------------------------------
// Kernel 2: edge scatter-add  agg[dst[e]] += x[src[e]]
// 16 threads per edge, each owns 4 consecutive features (float4 gather,
// 4 hardware global_atomic_add_f32). Both x and agg fit in the 192MB L2,
// so this is L2-atomic bound; unsafeAtomicAdd avoids CAS loops.
// ---------------------------------------------------------------------------
__global__ void gcn_scatter_kernel(const float* __restrict__ x,
                                   const int* __restrict__ src,
                                   const int* __restrict__ dst,
                                   float* __restrict__ agg,
                                   int n_edges) {
    int tid = blockIdx.x * blockDim.x + threadIdx.x;
    int e = tid >> 4;
    if (e >= n_edges) return;
    int d = (tid & 15) << 2;
    int s = src[e];
    int t = dst[e];
    const float4 v = *(const float4*)(x + (long)s * D + d);
    float* o = agg + (long)t * D + d;
    unsafeAtomicAdd(o + 0, v.x);
    unsafeAtomicAdd(o + 1, v.y);
    unsafeAtomicAdd(o + 2, v.z);
    unsafeAtomicAdd(o + 3, v.w);
}

// ---------------------------------------------------------------------------
// Kernel 3: out = relu(x @ U^T + agg @ V^T) via V_WMMA_F32_16X16X4_F32.
// One wave computes one 16x16 output tile; K=64 -> 16 WMMA steps per GEMM,
// both GEMMs accumulate into the same v8f accumulator, then fused ReLU.
//
// fp32 A-fragment (16x4) per-lane layout (ISA 7.12.2):
//   h = lane>>4, m = rowBase + (lane&15):  a = { A[m][k0+2h], A[m][k0+2h+1] }
// fp32 B-fragment (4x16): same index pattern on B[k][n] = U[n][k], i.e. we
// read rows of U/V (stored [d_out][d_in], row-contiguous).
// C/D (16x16, 8 VGPRs): acc[r] is row rowBase + r + 8h, col colBase+(lane&15).
// ---------------------------------------------------------------------------
__global__ void gcn_wmma_relu_kernel(const float* __restrict__ x,
                                     const float* __restrict__ agg,
                                     const float* __restrict__ U,
                                     const float* __restrict__ V,
                                     float* __restrict__ out,
                                     int n_tiles_m) {
    int wave = (blockIdx.x * blockDim.x + threadIdx.x) >> 5;
    if (wave >= n_tiles_m * 4) return;  // uniform per wave: EXEC stays all-1s
    int lane   = threadIdx.x & 31;
    int lane16 = lane & 15;
    int half   = lane >> 4;

    int mTile = wave >> 2;
    int nTile = wave & 3;
    int rowBase = mTile * 16;
    int col     = nTile * 16 + lane16;   // output column owned by this lane
    int mA      = rowBase + lane16;      // A row owned by this lane

    const float* xrow = x   + (long)mA * D;
    const float* grow = agg + (long)mA * D;
    const float* urow = U   + (long)col * D;  // B[k][n] = U[n][k]
    const float* vrow = V   + (long)col * D;

    v8f acc = {};

    // out += x @ U^T
#pragma unroll
    for (int k0 = 0; k0 < D; k0 += 4) {
        int kk = k0 + 2 * half;
        v2f a = { xrow[kk], xrow[kk + 1] };
        v2f b = { urow[kk], urow[kk + 1] };
        acc = __builtin_amdgcn_wmma_f32_16x16x4_f32(
            /*neg_a=*/false, a, /*neg_b=*/false, b,
            /*c_mod=*/(short)0, acc, /*reuse_a=*/false, /*reuse_b=*/false);
    }
    // out += agg @ V^T
#pragma unroll
    for (int k0 = 0; k0 < D; k0 += 4) {
        int kk = k0 + 2 * half;
        v2f a = { grow[kk], grow[kk + 1] };
        v2f b = { vrow[kk], vrow[kk + 1] };
        acc = __builtin_amdgcn_wmma_f32_16x16x4_f32(
            false, a, false, b, (short)0, acc, false, false);
    }

    // ReLU + store: acc[r] -> out[rowBase + r + 8*half][col]
#pragma unroll
    for (int r = 0; r < 8; ++r) {
        int m = rowBase + r + 8 * half;
        float v = acc[r];
        out[(long)m * D + col] = v > 0.f ? v : 0.f;
    }
}

// ---------------------------------------------------------------------------
extern "C" void kernel_launch(void* const* d_in, const int* in_sizes, int n_in,
                              void* d_out, int out_size, void* d_ws, size_t ws_size,
                              hipStream_t stream) {
    const float* x   = (const float*)d_in[0];
    const int*   src = (const int*)d_in[1];
    const int*   dst = (const int*)d_in[2];
    const float* U   = (const float*)d_in[3];
    const float* V   = (const float*)d_in[4];
    float*       out = (float*)d_out;

    int n_nodes = in_sizes[0] / D;   // 50000
    int n_edges = in_sizes[1];       // 1000000

    float* agg = (float*)d_ws;       // n_nodes * D floats (~12.8 MB)

    // 1) zero agg
    int n4 = n_nodes * (D / 4);
    gcn_zero_kernel<<<(n4 + 255) / 256, 256, 0, stream>>>((float4*)agg, n4);

    // 2) scatter-add over edges (16 threads per edge)
    long sthreads = (long)n_edges * 16;
    gcn_scatter_kernel<<<(int)((sthreads + 255) / 256), 256, 0, stream>>>(
        x, src, dst, agg, n_edges);

    // 3) fused dual-GEMM + ReLU with fp32 WMMA (one wave per 16x16 tile)
    int mTiles = n_nodes / 16;       // 3125 (exact)
    int waves  = mTiles * 4;         // 12500
    gcn_wmma_relu_kernel<<<(waves + 3) / 4, 128, 0, stream>>>(
        x, agg, U, V, out, mTiles);
}